// PredictionHead1D_16458314678406
// MI455X (gfx1250) — compile-verified
//
#include <hip/hip_runtime.h>

// CDNA5 / gfx1250: rank-4 log-space GEMM formulation of max-pooled Gaussians.
// Z[p,q] = A[p,:] (1x4) dot B[:,q] (4x1)  ==  -(d^2)*inv2s2 for point p, pixel q.
// Computed with V_WMMA_F32_16X16X4_F32 (K=4 matches the rank exactly), max-reduced
// over P tiles, then one exp per output pixel.

typedef __attribute__((ext_vector_type(2))) float v2f;
typedef __attribute__((ext_vector_type(8))) float v8f;

#define HH 96
#define WW 96
#define PP (HH * WW)          // 9216 points / pixels per image
#define SEG_THR 0.7f
#define GAUSS_THR 0.7f
#define EPSF 1e-7f
#define NEG_BIG -1e30f
#define CENTER 47.5f          // (96-1)/2, shrink coordinate magnitudes

// grid = (B, 12), block = 256 threads (8 waves, wave32).
// Dynamic LDS: PP*4 floats = 147456 B (fits CDNA5's 320 KB WGP LDS).
__global__ void gauss_maxpool_wmma(const float* __restrict__ var,
                                   const float* __restrict__ seg,
                                   float* __restrict__ out)
{
    extern __shared__ float As[];            // A matrix, row-major P x 4
    const int img = blockIdx.x;
    const float* v = var + img * PP;
    const float* s = seg + img * PP;
    float* o = out + img * PP;

    // ---- Stage A coefficients for the whole image into LDS ----
    for (int p = threadIdx.x; p < PP; p += blockDim.x) {
        float vv = v[p];
        float ss = s[p];
        bool valid = (ss > SEG_THR) && (vv > 0.0f);
        float x = (float)(p / WW) - CENTER;  // row coord (height axis)
        float y = (float)(p % WW) - CENTER;  // col coord (width axis)
        float a0, a1, a2, a3;
        if (valid) {
            float t = vv + EPSF;
            float inv = 1.0f / (2.0f * t * t);   // inv2s2
            a0 = -inv;
            a1 = 2.0f * inv * x;
            a2 = 2.0f * inv * y;
            a3 = -inv * (x * x + y * y);
        } else {
            a0 = a1 = a2 = 0.0f;
            a3 = NEG_BIG;                        // never wins the max
        }
        As[p * 4 + 0] = a0;
        As[p * 4 + 1] = a1;
        As[p * 4 + 2] = a2;
        As[p * 4 + 3] = a3;
    }
    __syncthreads();

    const int lane = threadIdx.x & 31;
    const int wave = threadIdx.x >> 5;
    const int gw = blockIdx.y * 8 + wave;    // 0..95 waves per image
    const int n = lane & 15;                 // column within 16-wide tile
    const int hi = lane >> 4;                // 0: K=0,1   1: K=2,3 (A/B frag halves)

    const float2* As2 = (const float2*)As;   // 2 float2 per point row

    // Each wave owns 6 Q-tiles (6*96 = 576 = PP/16 tiles per image)
    for (int k = 0; k < 6; ++k) {
        int qt = gw * 6 + k;                 // 0..575
        int i  = qt / (WW / 16);             // pixel row
        int j0 = (qt % (WW / 16)) * 16;      // first pixel column of tile
        float ic = (float)i - CENTER;
        float jc = (float)(j0 + n) - CENTER;

        // B fragment (4x16 f32): VGPR0 = rows {K=0 | K=2}, VGPR1 = rows {K=1 | K=3}
        v2f bfrag;
        if (hi == 0) { bfrag.x = ic * ic + jc * jc; bfrag.y = ic;   }
        else         { bfrag.x = jc;                bfrag.y = 1.0f; }

        v8f m;
        #pragma unroll
        for (int r = 0; r < 8; ++r) m[r] = NEG_BIG;
        v8f cz = {};                         // zero accumulator

        // Stream all P tiles: 1 WMMA (full K) + 8 v_max per 16x16 tile
        for (int t = 0; t < PP / 16; ++t) {
            // A fragment (16x4 f32): lane -> point row t*16 + n, K-pair selected by hi
            float2 af = As2[(t * 16 + n) * 2 + hi];   // ds_load_b64
            v2f afrag; afrag.x = af.x; afrag.y = af.y;
            v8f d = __builtin_amdgcn_wmma_f32_16x16x4_f32(
                false, afrag, false, bfrag, (short)0, cz, false, false);
            #pragma unroll
            for (int r = 0; r < 8; ++r) m[r] = fmaxf(m[r], d[r]);
        }

        // Column max: D layout -> 8 rows per lane, rows 8..15 in lanes 16..31
        float rmax = m[0];
        #pragma unroll
        for (int r = 1; r < 8; ++r) rmax = fmaxf(rmax, m[r]);
        rmax = fmaxf(rmax, __shfl_xor(rmax, 16, 32));

        // Inject exact z=0 self term for valid pixels; clamp (true z <= 0 always)
        int q = i * WW + j0 + n;
        float vv = v[q], ss = s[q];
        bool valid = (ss > SEG_THR) && (vv > 0.0f);
        if (valid) rmax = fmaxf(rmax, 0.0f);
        rmax = fminf(rmax, 0.0f);

        float pooled = __expf(rmax + EPSF);
        float res = (pooled >= GAUSS_THR) ? pooled : 0.0f;
        if (hi == 0) o[q] = res;             // lanes 0..15 hold the 16 columns
    }
}

extern "C" void kernel_launch(void* const* d_in, const int* in_sizes, int n_in,
                              void* d_out, int out_size, void* d_ws, size_t ws_size,
                              hipStream_t stream) {
    const float* var = (const float*)d_in[0];   // variance_map [B,1,96,96]
    const float* seg = (const float*)d_in[1];   // segmentation_map [B,1,96,96]
    float* out = (float*)d_out;                 // [B,1,96,96]
    int B = in_sizes[0] / PP;                   // 2
    dim3 grid(B, 12);
    dim3 block(256);
    size_t lds_bytes = (size_t)PP * 4 * sizeof(float);  // 147456 B
    gauss_maxpool_wmma<<<grid, block, lds_bytes, stream>>>(var, seg, out);
}